// PatchExtractor_11725260718482
// MI455X (gfx1250) — compile-verified
//
#include <hip/hip_runtime.h>

// Problem geometry (matches reference exactly)
#define B_    32
#define HH    512
#define WW    512
#define CC    3
#define PP    16
#define NP    1024                 // patches per image (32*32)
#define ROWF  (WW * CC)            // 1536 floats per image row
#define PATF  (PP * PP * CC)       // 768 floats per patch
#define PROW  (PP * CC)            // 48 floats per patch row

// b128 async ops take pointers to 4xi32 vectors in global (AS1) / LDS (AS3)
typedef int v4i __attribute__((vector_size(16)));
typedef __attribute__((address_space(1))) v4i gv4i_t;   // global
typedef __attribute__((address_space(3))) v4i lv4i_t;   // LDS

// TDM descriptor SGPR groups (per probe: g0=uint32x4, g1=int32x8, g2/g3=int32x4)
typedef unsigned int u32x4 __attribute__((vector_size(16)));
typedef int          i32x4 __attribute__((vector_size(16)));
typedef int          i32x8 __attribute__((vector_size(32)));

// ---------------------------------------------------------------------------
// K1: per-patch "any element > 0" mask. One wave32 per patch, 8 waves/block.
// Each lane: 6 x float4 loads (16B-aligned; 48-float rows => a 4-float chunk
// never straddles a row). Incremental (r,c) update: +128 floats = +2 rows +32
// cols (one divide per lane instead of six). Reduce with wave32 ballot.
// ---------------------------------------------------------------------------
__global__ __launch_bounds__(256) void k_mask(const float* __restrict__ img,
                                              int* __restrict__ mask) {
  const int wave = threadIdx.x >> 5;
  const int lane = threadIdx.x & 31;
  const int gp   = blockIdx.x * 8 + wave;        // global patch id in [0, B*NP)
  const int b    = gp >> 10;
  const int n    = gp & (NP - 1);
  const int ph   = n >> 5;
  const int pw   = n & 31;
  const float* base = img + ((size_t)b * HH + (size_t)ph * PP) * ROWF + pw * PROW;

  // CDNA5 speculative prefetch of the patch rows (global_prefetch_b8)
  __builtin_prefetch(base + (size_t)(lane & 15) * ROWF, 0, 0);

  int r = (lane * 4) / PROW;
  int c = (lane * 4) - r * PROW;
  bool pred = false;
#pragma unroll
  for (int i = 0; i < 6; ++i) {
    const float4 v = *(const float4*)(base + (size_t)r * ROWF + c);
    pred = pred | (v.x > 0.0f) | (v.y > 0.0f) | (v.z > 0.0f) | (v.w > 0.0f);
    c += 32; r += 2;
    if (c >= PROW) { c -= PROW; r += 1; }
  }

#if __has_builtin(__builtin_amdgcn_ballot_w32)
  const unsigned bal = __builtin_amdgcn_ballot_w32(pred);
  if (lane == 0) mask[gp] = (bal != 0u) ? 1 : 0;
#else
  const int any = __any(pred);
  if (lane == 0) mask[gp] = any ? 1 : 0;
#endif
}

// ---------------------------------------------------------------------------
// K2: per-image stable compaction map. One 1024-thread block per image.
// Inclusive Hillis-Steele scan of mask bits in LDS; kept patch n with
// exclusive prefix p becomes source of output slot p. Slots >= count get -1.
// ---------------------------------------------------------------------------
__global__ __launch_bounds__(1024) void k_scan(const int* __restrict__ mask,
                                               int* __restrict__ src_of) {
  __shared__ int sc[NP];
  __shared__ int so[NP];
  const int b = blockIdx.x;
  const int t = threadIdx.x;

  const int m = mask[b * NP + t];
  sc[t] = m;
  __syncthreads();

  for (int off = 1; off < NP; off <<= 1) {
    const int v   = sc[t];
    const int add = (t >= off) ? sc[t - off] : 0;
    __syncthreads();
    sc[t] = v + add;
    __syncthreads();
  }

  const int excl = sc[t] - m;                    // stable exclusive prefix
  so[t] = -1;
  __syncthreads();
  if (m) so[excl] = t;
  __syncthreads();

  src_of[b * NP + t] = so[t];
}

// ---------------------------------------------------------------------------
// K3: compacting copy on the CDNA5 DMA paths.
// One wave32 per OUTPUT slot, 8 waves/block, private 3KB LDS stage per wave.
// Kept patch:
//   TDM: one tensor_load_to_lds with a 2D D# (tile 48x16 f32, tensor row
//   stride 1536 f32) pulls the whole 3KB patch global->LDS in ONE
//   instruction; s_wait_tensorcnt 0; then 6x global_store_async_from_lds_b128
//   stream LDS->global (contiguous output tile) on the ASYNCcnt path.
//   Descriptor words are wave-uniform (one patch per wave) and forced through
//   readfirstlane so the SGPR-group operands are scalar.
// Empty slot: coalesced float4 zero stores.
// Input re-read hits L2 (100.7MB < 192MB, resident from K1).
// ---------------------------------------------------------------------------
__global__ __launch_bounds__(256) void k_gather(const float* __restrict__ img,
                                                const int* __restrict__ src_of,
                                                float* __restrict__ out) {
  __shared__ float stage[8 * PATF];              // 24 KB of 320 KB LDS
  const int wave = threadIdx.x >> 5;
  const int lane = threadIdx.x & 31;
  const int gp   = blockIdx.x * 8 + wave;        // output slot id
  const int b    = gp >> 10;
  const int src  = src_of[gp];
  float* outp    = out + (size_t)gp * PATF;

  if (src >= 0) {
    const int ph = src >> 5;
    const int pw = src & 31;
    const float* base =
        img + ((size_t)b * HH + (size_t)ph * PP) * ROWF + pw * PROW;
    float* sw = stage + wave * PATF;

#if __has_builtin(__builtin_amdgcn_tensor_load_to_lds) && \
    __has_builtin(__builtin_amdgcn_s_wait_tensorcnt)
    // ---- Tensor Data Mover load: whole patch tile in one instruction ----
    const unsigned long long ga =
        (unsigned long long)(__SIZE_TYPE__)base;           // global byte addr
    const unsigned lraw = (unsigned)(__SIZE_TYPE__)(lv4i_t*)sw; // LDS byte addr
    const unsigned ga_lo =
        (unsigned)__builtin_amdgcn_readfirstlane((int)(ga & 0xffffffffull));
    const unsigned ga_hi =
        (unsigned)__builtin_amdgcn_readfirstlane((int)(ga >> 32));
    const unsigned lds_u = (unsigned)__builtin_amdgcn_readfirstlane((int)lraw);

    // D# group0: [1:0]=count=1 | [63:32]=lds_addr | [120:64]=global_addr |
    //            [127:126]=type=2
    const u32x4 g0 = {1u, lds_u, ga_lo,
                      (ga_hi & 0x01ffffffu) | 0x80000000u};
    // D# group1: data_size=4B(code 2); tensor_dim0=48; tensor_dim1=16;
    //            tile_dim0=48; tile_dim1=16; tensor_dim0_stride=1536
    const i32x8 g1 = {(int)(2u << 16),     // w0: data_size in [17:16]
                      (int)(48u << 16),    // w1: tensor_dim0[15:0] @ [63:48]
                      (int)(16u << 16),    // w2: tensor_dim1[15:0] @ [95:80]
                      (int)(48u << 16),    // w3: tile_dim0 @ [127:112]
                      16,                  // w4: tile_dim1 @ [143:128]
                      1536,                // w5: tensor_dim0_stride[31:0]
                      0, 0};               // w6,w7: stride hi / dim1_stride
    const i32x4 gz = {0, 0, 0, 0};         // groups 2/3 unused (<=2D tensor)
#if defined(__clang_major__) && (__clang_major__ >= 23)
    const i32x8 gz8 = {0, 0, 0, 0, 0, 0, 0, 0};
    __builtin_amdgcn_tensor_load_to_lds(g0, g1, gz, gz, gz8, 0);
#else
    __builtin_amdgcn_tensor_load_to_lds(g0, g1, gz, gz, 0);
#endif
    __builtin_amdgcn_s_wait_tensorcnt(0);  // LDS tile ready
#else
    // ---- fallback: per-lane async b128 global->LDS ----
    int r = (lane * 4) / PROW;
    int c = (lane * 4) - r * PROW;
#pragma unroll
    for (int i = 0; i < 6; ++i) {
      const float* g = base + (size_t)r * ROWF + c;
      float* l = sw + (i * 32 + lane) * 4;
      __builtin_amdgcn_global_load_async_to_lds_b128((gv4i_t*)g, (lv4i_t*)l,
                                                     0, 0);
      c += 32; r += 2;
      if (c >= PROW) { c -= PROW; r += 1; }
    }
#if __has_builtin(__builtin_amdgcn_s_wait_asynccnt)
    __builtin_amdgcn_s_wait_asynccnt(0);
#else
    asm volatile("s_wait_asynccnt 0" ::: "memory");
#endif
#endif

    // ---- ASYNCcnt path: stream the staged tile LDS->global (contiguous) ----
#pragma unroll
    for (int i = 0; i < 6; ++i) {
      const int e = (i * 32 + lane) * 4;
      __builtin_amdgcn_global_store_async_from_lds_b128(
          (gv4i_t*)(outp + e), (lv4i_t*)(sw + e), 0, 0);
    }
    // S_ENDPGM performs an implicit wait-idle; async stores drain before exit.
  } else {
    const float4 z = make_float4(0.0f, 0.0f, 0.0f, 0.0f);
#pragma unroll
    for (int i = 0; i < 6; ++i) {
      *(float4*)(outp + (i * 32 + lane) * 4) = z;
    }
  }
}

// ---------------------------------------------------------------------------
extern "C" void kernel_launch(void* const* d_in, const int* in_sizes, int n_in,
                              void* d_out, int out_size, void* d_ws,
                              size_t ws_size, hipStream_t stream) {
  (void)in_sizes; (void)n_in; (void)out_size; (void)ws_size;
  const float* img = (const float*)d_in[0];
  float* out = (float*)d_out;

  int* mask   = (int*)d_ws;          //  B*NP ints = 128 KB
  int* src_of = mask + B_ * NP;      //  B*NP ints = 128 KB (256 KB total)

  const int patches = B_ * NP;       // 32768
  k_mask<<<patches / 8, 256, 0, stream>>>(img, mask);
  k_scan<<<B_, NP, 0, stream>>>(mask, src_of);
  k_gather<<<patches / 8, 256, 0, stream>>>(img, src_of, out);
}